// MultiGraphConvLayer_33139967656351
// MI455X (gfx1250) — compile-verified
//
#include <hip/hip_runtime.h>
#include <hip/hip_bf16.h>

// Graph scatter-add: out[dst] += concat(features[src], physics[e])
// Bandwidth-bound (~258 MB => ~11us floor @ 23.3 TB/s). No GEMM structure,
// so no WMMA; the CDNA5-specific paths used are the Tensor Data Mover
// (tensor_load_to_lds + s_wait_tensorcnt) for the streaming physics tile,
// global_prefetch_b8 for gather latency hiding, and f32 global atomics that
// resolve in the 192MB L2 (output is only 14.4MB).

#define N_NODES 50000
#define N_EDGES 800000
#define D_FEAT  64
#define D_PHYS  8
#define D_OUT   72

#define BLOCK   256                  // 8 waves (wave32)
#define EPB     512                  // edges per block (TDM tile rows)
#define WAVES   (BLOCK / 32)
#define EPW     (EPB / WAVES)        // 64 edges per wave

typedef __attribute__((ext_vector_type(4))) unsigned int u32x4;
typedef __attribute__((ext_vector_type(8))) int          i32x8;
typedef __attribute__((ext_vector_type(4))) int          i32x4;

__global__ __launch_bounds__(256) void zero_out_kernel(float* __restrict__ out, int n4) {
    int i      = blockIdx.x * blockDim.x + threadIdx.x;
    int stride = gridDim.x * blockDim.x;
    float4 z = make_float4(0.f, 0.f, 0.f, 0.f);
    for (int j = i; j < n4; j += stride)
        reinterpret_cast<float4*>(out)[j] = z;
}

__global__ __launch_bounds__(BLOCK) void
graphconv_scatter_kernel(const float*     __restrict__ features,
                         const long long* __restrict__ adj,       // [E][2] int64
                         const float*     __restrict__ phys,      // [E][8] f32
                         float*           __restrict__ out)       // [N][72] f32
{
    __shared__ float phys_tile[EPB * D_PHYS];   // 16 KB of the 320 KB/WGP LDS

    const int edge_base = blockIdx.x * EPB;
    const int tile_rows = min(EPB, N_EDGES - edge_base);

    // ---- TDM: DMA the contiguous (tile_rows x 8) f32 physics tile into LDS.
    // Issued by wave 0 only (TDM ignores EXEC; one issue per wave executing it).
    if (threadIdx.x < 32) {
        unsigned long long ga =
            (unsigned long long)(const void*)(phys + (size_t)edge_base * D_PHYS);
        unsigned int lds_off = (unsigned int)(size_t)&phys_tile[0];

        u32x4 g0;
        g0.x = 1u;                                         // count=1, user descriptor
        g0.y = lds_off;                                    // lds_addr (bytes)
        g0.z = (unsigned int)(ga & 0xFFFFFFFFu);           // global_addr[31:0]
        g0.w = (unsigned int)((ga >> 32) & 0x01FFFFFFu)    // global_addr[56:32]
             | (2u << 30);                                 // type=2 ("image")

        i32x8 g1;
        g1.s0 = (int)(2u << 16);                           // data_size=2 (4B); mask=0
        g1.s1 = (int)((unsigned)D_PHYS << 16);             // tensor_dim0 = 8 (lo16)
        g1.s2 = (int)(((unsigned)N_EDGES & 0xFFFFu) << 16);// dim0 hi=0 | tensor_dim1 lo16
        g1.s3 = (int)((((unsigned)N_EDGES >> 16) & 0xFFFFu)// tensor_dim1 hi16
             | ((unsigned)D_PHYS << 16));                  // tile_dim0 = 8
        g1.s4 = tile_rows;                                 // tile_dim1; tile_dim2=0
        g1.s5 = D_PHYS;                                    // tensor_dim0_stride = 8
        g1.s6 = 0;                                         // stride hi | dim1_stride lo
        g1.s7 = 0;                                         // dim1_stride hi (unused, 2D)

        i32x4 z4 = {0, 0, 0, 0};                           // groups 2/3: 2-D tensor
        i32x8 z8 = {0, 0, 0, 0, 0, 0, 0, 0};               // extra group (clang-23 form)
        __builtin_amdgcn_tensor_load_to_lds(g0, g1, z4, z4, z8, 0);
        __builtin_amdgcn_s_wait_tensorcnt(0);
    }
    __syncthreads();

    const int wave    = threadIdx.x >> 5;
    const int lane    = threadIdx.x & 31;
    const int w_local = wave * EPW;                        // tile-local first edge

    #pragma unroll 2
    for (int i = 0; i < EPW; ++i) {
        const int e = edge_base + w_local + i;
        if (e >= N_EDGES) break;

        const int src = (int)adj[2 * (size_t)e];
        const int dst = (int)adj[2 * (size_t)e + 1];

        // Prefetch next edge's feature row: 32 lanes x 8B spans the 256B row.
        if (i + 1 < EPW && e + 1 < N_EDGES) {
            const int nsrc = (int)adj[2 * (size_t)(e + 1)];
            __builtin_prefetch(features + (size_t)nsrc * D_FEAT + lane * 2, 0, 0);
        }

        // Gather: lane l takes dims (2l, 2l+1) -> one b64 load, full 256B/row/wave.
        const float2 f =
            *reinterpret_cast<const float2*>(features + (size_t)src * D_FEAT + lane * 2);

        float* orow = out + (size_t)dst * D_OUT;
        atomicAdd(orow + 2 * lane,     f.x);   // global_atomic_add_f32 (no return)
        atomicAdd(orow + 2 * lane + 1, f.y);
        if (lane < D_PHYS)
            atomicAdd(orow + D_FEAT + lane,
                      phys_tile[(w_local + i) * D_PHYS + lane]);
    }
}

extern "C" void kernel_launch(void* const* d_in, const int* in_sizes, int n_in,
                              void* d_out, int out_size, void* d_ws, size_t ws_size,
                              hipStream_t stream) {
    const float*     features = (const float*)d_in[0];
    const long long* adj      = (const long long*)d_in[1];   // int64 adjacency
    const float*     phys     = (const float*)d_in[2];
    float*           out      = (float*)d_out;               // 50000*72 f32

    // Zero the (poisoned) output: 3.6M floats = 900K float4 stores.
    const int n4 = (N_NODES * D_OUT) / 4;
    zero_out_kernel<<<1024, 256, 0, stream>>>(out, n4);

    // Scatter-add over edges.
    const int nblocks = (N_EDGES + EPB - 1) / EPB;           // 1563
    graphconv_scatter_kernel<<<nblocks, BLOCK, 0, stream>>>(features, adj, phys, out);
}